// STARNextItNet_20426864460415
// MI455X (gfx1250) — compile-verified
//
#include <hip/hip_runtime.h>

typedef __attribute__((ext_vector_type(2))) float v2f;
typedef __attribute__((ext_vector_type(8))) float v8f;

constexpr int kB    = 1024;
constexpr int kL    = 200;
constexpr int kD    = 128;
constexpr int kITEM = 100000;
constexpr int kSE   = 32;
constexpr int kSH   = 128;
constexpr int kDH   = 64;

// ---------------------------------------------------------------------------
// 1) masked mean-pool: mean[b][d] = sum_{l<len} item_emb[seq[l]][d] / len
// ---------------------------------------------------------------------------
__global__ void meanpool_kernel(const int* __restrict__ inputs,
                                const int* __restrict__ len_state,
                                const float* __restrict__ item_emb,
                                float* __restrict__ mean) {
    const int b   = blockIdx.x;
    const int tid = threadIdx.x;                 // 0..127
    const int* seq = inputs + (long)b * kL;
    const int len  = len_state[b];
    float acc = 0.f;
    int l = 0;
    for (; l + 4 <= len; l += 4) {
        int i0 = seq[l + 0], i1 = seq[l + 1], i2 = seq[l + 2], i3 = seq[l + 3];
        acc += item_emb[(long)i0 * kD + tid];
        acc += item_emb[(long)i1 * kD + tid];
        acc += item_emb[(long)i2 * kD + tid];
        acc += item_emb[(long)i3 * kD + tid];
    }
    for (; l < len; ++l)
        acc += item_emb[(long)seq[l] * kD + tid];
    mean[(long)b * kD + tid] = acc / (float)len;
}

// ---------------------------------------------------------------------------
// 2) generic fp32 WMMA GEMM (single 16x16 tile per wave) for the small layers:
//    C[M,N](ldc) = [relu](A[M,K](lda) @ W[K,N](ldw) + bias[N])
//    Marching pointers: loads use constant/small offsets, 3 pointer adds/step.
// ---------------------------------------------------------------------------
__global__ void gemm_wmma_f32(const float* __restrict__ A, int lda,
                              const float* __restrict__ W, int ldw,
                              const float* __restrict__ bias,
                              float* __restrict__ C, int ldc,
                              int M, int N, int K, int do_relu) {
    const int lane = threadIdx.x & 31;
    const int wave = threadIdx.x >> 5;
    const int tiles_n = N >> 4;
    const int tiles_m = M >> 4;
    const int tile = blockIdx.x * (blockDim.x >> 5) + wave;
    if (tile >= tiles_m * tiles_n) return;       // wave-uniform guard: EXEC stays all-ones
    const int tm = tile / tiles_n;
    const int tn = tile - tm * tiles_n;

    const int mrow = (tm << 4) + (lane & 15);
    const int ncol = (tn << 4) + (lane & 15);
    const int kb   = (lane >> 4) << 1;           // 0 or 2

    v8f acc;
    const float bv = bias[ncol];
#pragma unroll
    for (int i = 0; i < 8; ++i) acc[i] = bv;

    const float* __restrict__ ap  = A + (long)mrow * lda + kb;
    const float* __restrict__ wp0 = W + (long)kb * ldw + ncol;
    const float* __restrict__ wp1 = wp0 + ldw;
    const long   wstep = (long)4 * ldw;
    for (int k0 = 0; k0 < K; k0 += 4) {
        v2f a, bfr;
        a.x   = ap[0];
        a.y   = ap[1];
        bfr.x = wp0[0];
        bfr.y = wp1[0];
        acc = __builtin_amdgcn_wmma_f32_16x16x4_f32(
            false, a, false, bfr, (short)0, acc, false, false);
        ap  += 4;
        wp0 += wstep;
        wp1 += wstep;
    }

    const int rbase = (tm << 4) + ((lane >> 4) << 3);
    float* __restrict__ cp = C + (long)rbase * ldc + ncol;
#pragma unroll
    for (int v = 0; v < 8; ++v) {
        float r = acc[v];
        if (do_relu) r = fmaxf(r, 0.f);
        cp[(long)v * ldc] = r;
    }
}

// ---------------------------------------------------------------------------
// 3) dominant GEMM, register-blocked: out[1024,100000] = y[1024,128] @ cW3 + cB3
//    Each wave computes an MT(4) x NT(5) super-tile (64 rows x 80 cols).
//    Per K-step: 4 A b64 + 10 B b32 loads (all immediate-offset off two
//    marching pointers) feed 20 WMMAs. A reused NTx, B reused MTx in regs.
// ---------------------------------------------------------------------------
constexpr int MT = 4;
constexpr int NT = 5;

__global__ void gemm_big_wmma(const float* __restrict__ A,    // [1024,128]
                              const float* __restrict__ W,    // [128,100000]
                              const float* __restrict__ bias, // [100000]
                              float* __restrict__ C) {        // [1024,100000]
    constexpr int M = kB, N = kITEM, K = kD;
    constexpr int GM = M / (16 * MT);      // 16 groups along M
    constexpr int GN = N / (16 * NT);      // 1250 groups along N
    const int lane = threadIdx.x & 31;
    const int wave = threadIdx.x >> 5;
    const int grp  = blockIdx.x * (blockDim.x >> 5) + wave;
    if (grp >= GM * GN) return;            // wave-uniform
    const int gm = grp / GN;
    const int gn = grp - gm * GN;
    const int m0 = gm * (16 * MT);
    const int n0 = gn * (16 * NT);
    const int lm = lane & 15;
    const int kb = (lane >> 4) << 1;       // 0 or 2

    v8f acc[MT][NT];
#pragma unroll
    for (int nt = 0; nt < NT; ++nt) {
        const float bv = bias[n0 + nt * 16 + lm];
#pragma unroll
        for (int mt = 0; mt < MT; ++mt)
#pragma unroll
            for (int v = 0; v < 8; ++v) acc[mt][nt][v] = bv;
    }

    // marching pointers; every load below is base + compile-time immediate
    const float* __restrict__ ap = A + (long)(m0 + lm) * K + kb;   // += 4 / step
    const float* __restrict__ wp = W + (long)kb * N + n0 + lm;     // += 4*N / step
    for (int k0 = 0; k0 < K; k0 += 4) {
        v2f a[MT];
#pragma unroll
        for (int mt = 0; mt < MT; ++mt) {
            a[mt].x = ap[mt * 16 * K];         // 8*mt KB immediate
            a[mt].y = ap[mt * 16 * K + 1];
        }
        v2f bf[NT];
#pragma unroll
        for (int nt = 0; nt < NT; ++nt) {
            bf[nt].x = wp[nt * 16];            // 64*nt B immediate
            bf[nt].y = wp[nt * 16 + N];        // +400 KB immediate (fits imm24)
        }
#pragma unroll
        for (int mt = 0; mt < MT; ++mt)
#pragma unroll
            for (int nt = 0; nt < NT; ++nt)
                acc[mt][nt] = __builtin_amdgcn_wmma_f32_16x16x4_f32(
                    false, a[mt], false, bf[nt], (short)0, acc[mt][nt], false, false);
        ap += 4;
        wp += (long)4 * N;
    }

    const int rb = (lane >> 4) << 3;       // +8 rows for upper half-wave
#pragma unroll
    for (int mt = 0; mt < MT; ++mt) {
        float* __restrict__ cp = C + (long)(m0 + mt * 16 + rb) * N + n0 + lm;
#pragma unroll
        for (int v = 0; v < 8; ++v)
#pragma unroll
            for (int nt = 0; nt < NT; ++nt)
                cp[(long)v * N + nt * 16] = acc[mt][nt][v];   // max 2.8 MB immediate
    }
}

// ---------------------------------------------------------------------------
// 4) scenario-routed expert branch (per-row gathered weights, VALU):
//    dom written into comb[:, 128:192] (row stride 192).
// ---------------------------------------------------------------------------
__global__ void expert_kernel(const float* __restrict__ mean,
                              const int*   __restrict__ scenario_ids,
                              const float* __restrict__ scen_emb,
                              const float* __restrict__ pW, const float* __restrict__ pB,
                              const float* __restrict__ dW1, const float* __restrict__ dB1,
                              const float* __restrict__ dW2, const float* __restrict__ dB2,
                              float* __restrict__ comb) {
    const int b   = blockIdx.x;
    const int tid = threadIdx.x;                 // 0..127
    const int sid = scenario_ids[b];

    __shared__ float ses[kSE];
    __shared__ float xs[kD];
    __shared__ float hs[kDH];

    if (tid < kSE) ses[tid] = scen_emb[(long)sid * kSE + tid];
    __syncthreads();

    float acc = pB[(long)sid * kD + tid];
    const float* pWs = pW + (long)sid * kSE * kD;
#pragma unroll 8
    for (int e = 0; e < kSE; ++e)
        acc += ses[e] * pWs[(long)e * kD + tid];
    xs[tid] = mean[(long)b * kD + tid] + acc;
    __syncthreads();

    if (tid < kDH) {
        float h = dB1[(long)sid * kDH + tid];
        const float* w = dW1 + (long)sid * kD * kDH;
#pragma unroll 4
        for (int k = 0; k < kD; ++k)
            h += xs[k] * w[(long)k * kDH + tid];
        hs[tid] = fmaxf(h, 0.f);
    }
    __syncthreads();

    if (tid < kDH) {
        float d2 = dB2[(long)sid * kDH + tid];
        const float* w = dW2 + (long)sid * kDH * kDH;
#pragma unroll 4
        for (int k = 0; k < kDH; ++k)
            d2 += hs[k] * w[(long)k * kDH + tid];
        comb[(long)b * (kSH + kDH) + kSH + tid] = fmaxf(d2, 0.f);
    }
}

// ---------------------------------------------------------------------------
// launch
// ---------------------------------------------------------------------------
extern "C" void kernel_launch(void* const* d_in, const int* in_sizes, int n_in,
                              void* d_out, int out_size, void* d_ws, size_t ws_size,
                              hipStream_t stream) {
    const int*   inputs    = (const int*)  d_in[0];
    const int*   len_state = (const int*)  d_in[1];
    const int*   scen_ids  = (const int*)  d_in[2];
    const float* item_emb  = (const float*)d_in[3];
    const float* scen_emb  = (const float*)d_in[4];
    const float* sW1 = (const float*)d_in[5];
    const float* sB1 = (const float*)d_in[6];
    const float* sW2 = (const float*)d_in[7];
    const float* sB2 = (const float*)d_in[8];
    const float* pW  = (const float*)d_in[9];
    const float* pB  = (const float*)d_in[10];
    const float* dW1 = (const float*)d_in[11];
    const float* dB1 = (const float*)d_in[12];
    const float* dW2 = (const float*)d_in[13];
    const float* dB2 = (const float*)d_in[14];
    const float* fW1 = (const float*)d_in[15];
    const float* fB1 = (const float*)d_in[16];
    const float* fW2 = (const float*)d_in[17];
    const float* fB2 = (const float*)d_in[18];
    const float* cW1 = (const float*)d_in[19];
    const float* cB1 = (const float*)d_in[20];
    const float* cW2 = (const float*)d_in[21];
    const float* cB2 = (const float*)d_in[22];
    const float* cW3 = (const float*)d_in[23];
    const float* cB3 = (const float*)d_in[24];
    float* out = (float*)d_out;

    float* ws    = (float*)d_ws;
    float* mean  = ws; ws += kB * kD;            // [B,128]
    float* t1    = ws; ws += kB * kSH;           // shared layer1   [B,128]
    float* comb  = ws; ws += kB * (kSH + kDH);   // [B,192]: shared2 | dom
    float* f1    = ws; ws += kB * kD;            // relu(comb@fW1)  [B,128]
    float* fused = ws; ws += kB * kD;            // fusion output   [B,128]
    float* h1    = ws; ws += kB * kD;            // head layer1     [B,128]
    float* y     = ws; ws += kB * kD;            // head layer2     [B,128]

    meanpool_kernel<<<kB, 128, 0, stream>>>(inputs, len_state, item_emb, mean);

    auto gemm = [&](const float* A, int lda, const float* W, int ldw,
                    const float* bias, float* Cp, int ldc,
                    int M, int N, int K, int relu) {
        const int wavesPerBlock = 8;
        const int tiles  = (M / 16) * (N / 16);
        const int blocks = (tiles + wavesPerBlock - 1) / wavesPerBlock;
        gemm_wmma_f32<<<blocks, 32 * wavesPerBlock, 0, stream>>>(
            A, lda, W, ldw, bias, Cp, ldc, M, N, K, relu);
    };

    // shared tower
    gemm(mean, kD,        sW1, kSH,   sB1, t1,   kSH,       kB, kSH, kD,        1);
    gemm(t1,   kSH,       sW2, kSH,   sB2, comb, kSH + kDH, kB, kSH, kSH,       1); // -> comb[:, :128]
    // expert branch -> comb[:, 128:192]
    expert_kernel<<<kB, 128, 0, stream>>>(mean, scen_ids, scen_emb,
                                          pW, pB, dW1, dB1, dW2, dB2, comb);
    // fusion
    gemm(comb, kSH + kDH, fW1, kD,    fB1, f1,    kD,       kB, kD,  kSH + kDH, 1);
    gemm(f1,   kD,        fW2, kD,    fB2, fused, kD,       kB, kD,  kD,        0);
    // classifier head
    gemm(fused, kD,       cW1, kD,    cB1, h1,    kD,       kB, kD,  kD,        1);
    gemm(h1,    kD,       cW2, kD,    cB2, y,     kD,       kB, kD,  kD,        1);
    // dominant GEMM: register-blocked WMMA, 4 waves/block
    {
        constexpr int groups = (kB / (16 * MT)) * (kITEM / (16 * NT)); // 16 * 1250
        const int wavesPerBlock = 4;
        const int blocks = (groups + wavesPerBlock - 1) / wavesPerBlock;
        gemm_big_wmma<<<blocks, 32 * wavesPerBlock, 0, stream>>>(y, cW3, cB3, out);
    }
}